// GIN_81784767250528
// MI455X (gfx1250) — compile-verified
//
#include <hip/hip_runtime.h>
#include <math.h>

#define N_NODES   100000
#define N_EDGES   1600000
#define F_DIM     64
#define N_CLASSES 10
#define N_GRAPHS  64
#define NODE_FEATS (N_NODES * F_DIM)

typedef float v2f __attribute__((ext_vector_type(2)));
typedef float v8f __attribute__((ext_vector_type(8)));

#define GEMM_WAVES   10
#define GEMM_THREADS (GEMM_WAVES * 32)
#define GEMM_BLOCKS  625            // 625 * 10 waves * 16 rows = 100000 exactly
#define WT_STRIDE    65             // pad to avoid 64-bank conflicts
#define ST_STRIDE    65

__device__ __forceinline__ float sigmoid_f(float v) {
    return 1.0f / (1.0f + __expf(-v));
}

// ---------------------------------------------------------------------------
// Edge scatter-add: agg[dst] += x[src].  4 threads per edge, 16 floats each.
// ---------------------------------------------------------------------------
__global__ __launch_bounds__(256)
void gin_scatter_kernel(const float* __restrict__ x, const int* __restrict__ ei,
                        float* __restrict__ agg)
{
    int t = blockIdx.x * 256 + threadIdx.x;
    int e = t >> 2;
    if (e >= N_EDGES) return;
    int c   = (t & 3) << 4;          // 0,16,32,48
    int src = ei[e];
    int dst = ei[N_EDGES + e];
    const float4* xs = (const float4*)(x + src * F_DIM + c);
    float* ad = agg + dst * F_DIM + c;
#pragma unroll
    for (int q = 0; q < 4; ++q) {
        float4 v = xs[q];
        unsafeAtomicAdd(ad + q * 4 + 0, v.x);
        unsafeAtomicAdd(ad + q * 4 + 1, v.y);
        unsafeAtomicAdd(ad + q * 4 + 2, v.z);
        unsafeAtomicAdd(ad + q * 4 + 3, v.w);
    }
}

// ---------------------------------------------------------------------------
// Fused GIN layer:  xout = sigmoid( sigmoid( (x+agg) @ W1^T ) @ W2^T )
// One wave per 16-row block, WMMA f32 16x16x4, weights transposed in LDS.
// xout may alias agg (each wave reads then writes only its own rows).
// ---------------------------------------------------------------------------
__global__ __launch_bounds__(GEMM_THREADS)
void gin_layer_kernel(const float* __restrict__ xcur, const float* __restrict__ agg,
                      const float* __restrict__ w1, const float* __restrict__ w2,
                      float* __restrict__ xout)
{
    __shared__ float wt[64 * WT_STRIDE];                     // 16.25 KB
    __shared__ float stage[GEMM_WAVES][16 * ST_STRIDE];      // 40.6 KB
    const int tid = threadIdx.x;

    // Load W1 transposed: wt[k][n] = w1[n*64+k]  (coalesced read, padded write)
    for (int i = tid; i < 4096; i += GEMM_THREADS) {
        int n = i >> 6, k = i & 63;
        wt[k * WT_STRIDE + n] = w1[i];
    }
    __syncthreads();

    const int wave    = tid >> 5;
    const int lane    = tid & 31;
    const int blk     = blockIdx.x * GEMM_WAVES + wave;      // always < 6250
    const int rowbase = blk * 16;
    float* st = stage[wave];

    // Stage A tile = x + agg (wave-private, coalesced)
    {
        const float* xp = xcur + rowbase * F_DIM;
        const float* ap = agg  + rowbase * F_DIM;
#pragma unroll
        for (int j = 0; j < 32; ++j) {
            int e = lane + 32 * j;
            int r = e >> 6, c = e & 63;
            st[r * ST_STRIDE + c] = xp[r * F_DIM + c] + ap[r * F_DIM + c];
        }
    }

    // WMMA operand lane mapping (wave32):
    //  A 16x4 f32 : lane&15 = M row, lane>>4 selects K pair {0,1} vs {2,3}
    //  B 4x16 f32 : lane&15 = N col, lane>>4 selects K pair
    //  C/D 16x16  : vgpr r -> M = r + 8*(lane>>4), N = lane&15 (+16*tile)
    const int mrow  = lane & 15;
    const int koff  = (lane >> 4) << 1;
    const int ncol  = lane & 15;
    const int mbase = (lane >> 4) << 3;

    v8f acc[4] = {{}, {}, {}, {}};
#pragma unroll
    for (int k0 = 0; k0 < 64; k0 += 4) {
        v2f a;
        a.x = st[mrow * ST_STRIDE + k0 + koff];
        a.y = st[mrow * ST_STRIDE + k0 + koff + 1];
#pragma unroll
        for (int tIdx = 0; tIdx < 4; ++tIdx) {
            v2f b;
            b.x = wt[(k0 + koff)     * WT_STRIDE + tIdx * 16 + ncol];
            b.y = wt[(k0 + koff + 1) * WT_STRIDE + tIdx * 16 + ncol];
            acc[tIdx] = __builtin_amdgcn_wmma_f32_16x16x4_f32(
                false, a, false, b, (short)0, acc[tIdx], false, false);
        }
    }

    // h = sigmoid(acc) back into the (now free) stage, in A-readable layout
#pragma unroll
    for (int tIdx = 0; tIdx < 4; ++tIdx)
#pragma unroll
        for (int r = 0; r < 8; ++r)
            st[(mbase + r) * ST_STRIDE + tIdx * 16 + ncol] = sigmoid_f(acc[tIdx][r]);

    __syncthreads();   // all waves done with W1
    for (int i = tid; i < 4096; i += GEMM_THREADS) {
        int n = i >> 6, k = i & 63;
        wt[k * WT_STRIDE + n] = w2[i];
    }
    __syncthreads();

    v8f acc2[4] = {{}, {}, {}, {}};
#pragma unroll
    for (int k0 = 0; k0 < 64; k0 += 4) {
        v2f a;
        a.x = st[mrow * ST_STRIDE + k0 + koff];
        a.y = st[mrow * ST_STRIDE + k0 + koff + 1];
#pragma unroll
        for (int tIdx = 0; tIdx < 4; ++tIdx) {
            v2f b;
            b.x = wt[(k0 + koff)     * WT_STRIDE + tIdx * 16 + ncol];
            b.y = wt[(k0 + koff + 1) * WT_STRIDE + tIdx * 16 + ncol];
            acc2[tIdx] = __builtin_amdgcn_wmma_f32_16x16x4_f32(
                false, a, false, b, (short)0, acc2[tIdx], false, false);
        }
    }

    float* op = xout + rowbase * F_DIM;
#pragma unroll
    for (int tIdx = 0; tIdx < 4; ++tIdx)
#pragma unroll
        for (int r = 0; r < 8; ++r)
            op[(mbase + r) * F_DIM + tIdx * 16 + ncol] = sigmoid_f(acc2[tIdx][r]);
}

// ---------------------------------------------------------------------------
// Global add-pool: pooled[g][f] += x[n][f] for batch[n]==g  (LDS partials)
// ---------------------------------------------------------------------------
__global__ __launch_bounds__(256)
void gin_pool_kernel(const float* __restrict__ x, const int* __restrict__ batch,
                     float* __restrict__ pooled)
{
    __shared__ float pl[N_GRAPHS * F_DIM];   // 16 KB
    const int tid = threadIdx.x;
    for (int i = tid; i < N_GRAPHS * F_DIM; i += 256) pl[i] = 0.0f;
    __syncthreads();

    const int f    = tid & 63;
    const int slot = tid >> 6;               // 0..3
    const int base = blockIdx.x * 2048;
    for (int i = 0; i < 512; ++i) {
        int node = base + slot + 4 * i;
        if (node < N_NODES) {
            int g = batch[node];
            atomicAdd(&pl[g * F_DIM + f], x[node * F_DIM + f]);
        }
    }
    __syncthreads();
    for (int i = tid; i < N_GRAPHS * F_DIM; i += 256)
        unsafeAtomicAdd(&pooled[i], pl[i]);
}

// ---------------------------------------------------------------------------
// Head: logits = pooled @ fc_w^T + fc_b ; log_softmax ; emit (logits, pooled)
// ---------------------------------------------------------------------------
__global__ __launch_bounds__(256)
void gin_head_kernel(const float* __restrict__ pooled, const float* __restrict__ fcw,
                     const float* __restrict__ fcb, float* __restrict__ out)
{
    const int tid = threadIdx.x;
    if (tid < N_GRAPHS) {
        float l[N_CLASSES];
        float m = -1e30f;
#pragma unroll
        for (int c = 0; c < N_CLASSES; ++c) {
            float s = fcb[c];
            for (int f = 0; f < F_DIM; ++f)
                s += pooled[tid * F_DIM + f] * fcw[c * F_DIM + f];
            l[c] = s;
            m = fmaxf(m, s);
        }
        float lse = 0.0f;
#pragma unroll
        for (int c = 0; c < N_CLASSES; ++c) lse += __expf(l[c] - m);
        lse = __logf(lse);
#pragma unroll
        for (int c = 0; c < N_CLASSES; ++c)
            out[tid * N_CLASSES + c] = l[c] - m - lse;
    }
    for (int i = tid; i < N_GRAPHS * F_DIM; i += 256)
        out[N_GRAPHS * N_CLASSES + i] = pooled[i];
}

// ---------------------------------------------------------------------------
extern "C" void kernel_launch(void* const* d_in, const int* in_sizes, int n_in,
                              void* d_out, int out_size, void* d_ws, size_t ws_size,
                              hipStream_t stream)
{
    const float* x     = (const float*)d_in[0];
    const int*   ei    = (const int*)d_in[1];
    const int*   batch = (const int*)d_in[2];
    const float* w1    = (const float*)d_in[3];   // [4][64][64]
    const float* w2    = (const float*)d_in[4];   // [4][64][64]
    const float* fcw   = (const float*)d_in[5];
    const float* fcb   = (const float*)d_in[6];
    float* out = (float*)d_out;

    float* bufA   = (float*)d_ws;
    float* bufB   = bufA + NODE_FEATS;
    float* pooled = bufB + NODE_FEATS;

    const int scatter_blocks = (N_EDGES * 4) / 256;   // 25000
    const size_t featBytes = (size_t)NODE_FEATS * sizeof(float);

    // layer 0: cur = input x, agg = bufA, next = bufB
    hipMemsetAsync(bufA, 0, featBytes, stream);
    gin_scatter_kernel<<<scatter_blocks, 256, 0, stream>>>(x, ei, bufA);
    gin_layer_kernel<<<GEMM_BLOCKS, GEMM_THREADS, 0, stream>>>(x, bufA, w1 + 0,    w2 + 0,    bufB);
    // layer 1: cur = bufB, agg = bufA, next = bufA (in-place over agg)
    hipMemsetAsync(bufA, 0, featBytes, stream);
    gin_scatter_kernel<<<scatter_blocks, 256, 0, stream>>>(bufB, ei, bufA);
    gin_layer_kernel<<<GEMM_BLOCKS, GEMM_THREADS, 0, stream>>>(bufB, bufA, w1 + 4096, w2 + 4096, bufA);
    // layer 2: cur = bufA, agg = bufB, next = bufB
    hipMemsetAsync(bufB, 0, featBytes, stream);
    gin_scatter_kernel<<<scatter_blocks, 256, 0, stream>>>(bufA, ei, bufB);
    gin_layer_kernel<<<GEMM_BLOCKS, GEMM_THREADS, 0, stream>>>(bufA, bufB, w1 + 8192, w2 + 8192, bufB);
    // layer 3: cur = bufB, agg = bufA, next = bufA
    hipMemsetAsync(bufA, 0, featBytes, stream);
    gin_scatter_kernel<<<scatter_blocks, 256, 0, stream>>>(bufB, ei, bufA);
    gin_layer_kernel<<<GEMM_BLOCKS, GEMM_THREADS, 0, stream>>>(bufB, bufA, w1 + 12288, w2 + 12288, bufA);

    // pooling + head
    hipMemsetAsync(pooled, 0, (size_t)N_GRAPHS * F_DIM * sizeof(float), stream);
    gin_pool_kernel<<<(N_NODES + 2047) / 2048, 256, 0, stream>>>(bufA, batch, pooled);
    gin_head_kernel<<<1, 256, 0, stream>>>(pooled, fcw, fcb, out);
}